// Connector_36026185679396
// MI455X (gfx1250) — compile-verified
//
#include <hip/hip_runtime.h>
#include <math.h>

#define TOK 8192      // B*S tokens
#define NN  4
#define CC  2048
#define KK  8192      // NN*CC
#define EPSF 1e-5f

#define KWAVES 4              // waves per gates-block (K-split factor)
#define KSLICE (KK / KWAVES)  // 2048 K per wave

typedef float v2f __attribute__((ext_vector_type(2)));
typedef float v4f __attribute__((ext_vector_type(4)));
typedef float v8f __attribute__((ext_vector_type(8)));

// ---------------------------------------------------------------------------
// Kernel A: pack alpha-scaled projection weights directly into WMMA B-fragment
// order:  fragB[k/4][lane][q], q=0..3  ==  { b0.x, b0.y, b1.x, b1.y } so the
// gates kernel loads each lane's whole B fragment with ONE global_load_b128.
//
//   lane = g*16 + n0           (g = half-wave, n0 = column-in-tile)
//   q=0 -> row 4*kb+2g+0, col n0        (N-tile 0)
//   q=1 -> row 4*kb+2g+1, col n0
//   q=2 -> row 4*kb+2g+0, col n0+16     (N-tile 1)
//   q=3 -> row 4*kb+2g+1, col n0+16
//
// Logical packed columns: 0-3 alpha_pre*phi_pre, 4-7 alpha_post*phi_post,
// 8-23 alpha_res*phi_res, 24-31 zero pad.
// ---------------------------------------------------------------------------
__global__ void __launch_bounds__(256)
pack_phi_kernel(const float* __restrict__ phi_pre,
                const float* __restrict__ phi_post,
                const float* __restrict__ phi_res,
                const float* __restrict__ alpha_pre,
                const float* __restrict__ alpha_post,
                const float* __restrict__ alpha_res,
                float* __restrict__ fragB) {
    int idx = blockIdx.x * blockDim.x + threadIdx.x;   // over KK*32 elements
    if (idx >= KK * 32) return;

    int q    = idx & 3;
    int lane = (idx >> 2) & 31;
    int kb   = idx >> 7;

    int g   = lane >> 4;
    int n0  = lane & 15;
    int row = 4 * kb + 2 * g + (q & 1);
    int col = n0 + 16 * (q >> 1);

    float v = 0.0f;
    if (col < 4)        v = alpha_pre[0]  * phi_pre[row * 4 + col];
    else if (col < 8)   v = alpha_post[0] * phi_post[row * 4 + (col - 4)];
    else if (col < 24)  v = alpha_res[0]  * phi_res[row * 16 + (col - 8)];
    fragB[idx] = v;
}

// ---------------------------------------------------------------------------
// Kernel B: per 16-token tile, 4 waves split the K=8192 reduction:
//   P[16][32] = X[16][8192] @ Phi[8192][32]   via V_WMMA_F32_16X16X4_F32
//   ss[16]    = rowwise sum of squares
// partials combined in LDS; wave 0 runs gates + 20-iter Sinkhorn epilogue.
//
// A-frag (16x4 f32): lanes 0-15 -> M=lane, V0=K+0, V1=K+1;
//                    lanes 16-31 -> M=lane-16, V0=K+2, V1=K+3.
// D-frag (16x16 f32): VGPR r: lanes 0-15 -> M=r, N=lane;
//                             lanes 16-31 -> M=r+8, N=lane-16.
// ---------------------------------------------------------------------------
__global__ void __launch_bounds__(KWAVES * 32)
gates_kernel(const float* __restrict__ x,
             const float* __restrict__ fragB,
             const float* __restrict__ b_pre,
             const float* __restrict__ b_post,
             const float* __restrict__ b_res,
             float* __restrict__ wHpre,
             float* __restrict__ wHpost,
             float* __restrict__ wM) {
    __shared__ float Pt[KWAVES][16][32];
    __shared__ float SSs[KWAVES][16];

    const int lane = threadIdx.x & 31;
    const int w    = threadIdx.x >> 5;       // wave id -> K slice
    const int mrow = lane & 15;
    const int g    = lane >> 4;
    const int t0   = blockIdx.x * 16;        // first token of tile

    const int kbeg = w * KSLICE;
    const int kend = kbeg + KSLICE;

    const float* xrow = x + (size_t)(t0 + mrow) * KK + 2 * g;
    const v4f*   bf   = (const v4f*)fragB + (size_t)(kbeg >> 2) * 32 + lane;

    v8f acc0 = {};   // N = 0..15
    v8f acc1 = {};   // N = 16..31
    float ss = 0.0f;

#pragma unroll 4
    for (int k = kbeg; k < kend; k += 4) {
        v2f a = *(const v2f*)(xrow + k);     // {x[k+2g], x[k+2g+1]}
        ss += a.x * a.x + a.y * a.y;

        v4f f = *bf;  bf += 32;              // whole B fragment: one b128
        v2f b0 = { f.x, f.y };
        v2f b1 = { f.z, f.w };

        acc0 = __builtin_amdgcn_wmma_f32_16x16x4_f32(false, a, false, b0,
                                                     (short)0, acc0, false, false);
        acc1 = __builtin_amdgcn_wmma_f32_16x16x4_f32(false, a, false, b1,
                                                     (short)0, acc1, false, false);
    }

    // rowwise sum-of-squares: lane L + lane L+16 cover a full row of the slice
    ss += __shfl_xor(ss, 16, 32);

    // spill partial P[16][32] using the C/D fragment layout
#pragma unroll
    for (int r = 0; r < 8; ++r) {
        Pt[w][r + 8 * g][mrow]      = acc0[r];
        Pt[w][r + 8 * g][mrow + 16] = acc1[r];
    }
    if (g == 0) SSs[w][mrow] = ss;
    __syncthreads();

    // epilogue: one token per lane (threads 0..15 of wave 0)
    if (threadIdx.x < 16) {
        const int l = threadIdx.x;
        const int t = t0 + l;

        float ssum = 0.0f;
#pragma unroll
        for (int u = 0; u < KWAVES; ++u) ssum += SSs[u][l];
        const float inv = rsqrtf(ssum * (1.0f / KK) + EPSF);   // 1/rms

        float P[32];
#pragma unroll
        for (int c = 0; c < 32; ++c) {
            float s = 0.0f;
#pragma unroll
            for (int u = 0; u < KWAVES; ++u) s += Pt[u][l][c];
            P[c] = s;
        }

        // gates (alpha folded into fragB at pack time)
#pragma unroll
        for (int j = 0; j < 4; ++j) {
            float tp = P[j]     * inv + b_pre[j];
            float tq = P[4 + j] * inv + b_post[j];
            wHpre[t * 4 + j]  = 1.0f / (1.0f + __expf(-tp));
            wHpost[t * 4 + j] = 2.0f / (1.0f + __expf(-tq));
        }

        // Sinkhorn-Knopp on 4x4
        float Mm[16];
#pragma unroll
        for (int e = 0; e < 16; ++e)
            Mm[e] = __expf(P[8 + e] * inv + b_res[e]);

        for (int it = 0; it < 20; ++it) {
#pragma unroll
            for (int i = 0; i < 4; ++i) {
                float s = Mm[4*i] + Mm[4*i+1] + Mm[4*i+2] + Mm[4*i+3];
                float r = 1.0f / (s + EPSF);
                Mm[4*i] *= r; Mm[4*i+1] *= r; Mm[4*i+2] *= r; Mm[4*i+3] *= r;
            }
#pragma unroll
            for (int j = 0; j < 4; ++j) {
                float s = Mm[j] + Mm[4+j] + Mm[8+j] + Mm[12+j];
                float r = 1.0f / (s + EPSF);
                Mm[j] *= r; Mm[4+j] *= r; Mm[8+j] *= r; Mm[12+j] *= r;
            }
        }
#pragma unroll
        for (int e = 0; e < 16; ++e) wM[t * 16 + e] = Mm[e];
    }
}

// ---------------------------------------------------------------------------
// Kernel C: streaming output pass. One 256-thread block per token.
//   x_agg = sum_j Hpre[j]*x[j,:]; RMSNorm over C; out = M@x + Hpost .* normed
// v4f loads, x[4][2048] in registers, nontemporal v4f stores.
// ---------------------------------------------------------------------------
__device__ __forceinline__ v4f v4fma(float a, v4f v, v4f acc) {
    acc.x += a * v.x; acc.y += a * v.y; acc.z += a * v.z; acc.w += a * v.w;
    return acc;
}
__device__ __forceinline__ float v4sq(v4f v) {
    return v.x*v.x + v.y*v.y + v.z*v.z + v.w*v.w;
}

__global__ void __launch_bounds__(256)
output_kernel(const float* __restrict__ x,
              const float* __restrict__ wHpre,
              const float* __restrict__ wHpost,
              const float* __restrict__ wM,
              const float* __restrict__ norm_scale,
              float* __restrict__ out) {
    const int t   = blockIdx.x;
    const int tid = threadIdx.x;

    __shared__ float sHpre[4], sHpost[4], sM[16];
    __shared__ float red[8];

    if (tid < 4)  { sHpre[tid] = wHpre[t*4 + tid]; sHpost[tid] = wHpost[t*4 + tid]; }
    if (tid < 16) { sM[tid] = wM[t*16 + tid]; }
    __syncthreads();

    const v4f* xf = (const v4f*)(x + (size_t)t * KK);   // [4][512] v4f

    v4f xv[4][2];
#pragma unroll
    for (int i = 0; i < 4; ++i) {
        xv[i][0] = xf[i * 512 + tid];
        xv[i][1] = xf[i * 512 + tid + 256];
    }

    // aggregate with Hpre
    v4f agg[2];
#pragma unroll
    for (int q = 0; q < 2; ++q) {
        v4f a = { 0.f, 0.f, 0.f, 0.f };
#pragma unroll
        for (int j = 0; j < 4; ++j) a = v4fma(sHpre[j], xv[j][q], a);
        agg[q] = a;
    }

    // block reduction of sum(agg^2) over C=2048 (8 wave32's)
    float s = v4sq(agg[0]) + v4sq(agg[1]);
#pragma unroll
    for (int off = 16; off > 0; off >>= 1) s += __shfl_xor(s, off, 32);
    if ((tid & 31) == 0) red[tid >> 5] = s;
    __syncthreads();
    float tot = 0.f;
#pragma unroll
    for (int u = 0; u < 8; ++u) tot += red[u];
    const float inv = rsqrtf(tot * (1.0f / CC) + EPSF);

    const v4f* nsf = (const v4f*)norm_scale;
    v4f nrm[2];
#pragma unroll
    for (int q = 0; q < 2; ++q) {
        v4f ns = nsf[tid + q * 256];
        nrm[q].x = agg[q].x * inv * ns.x;
        nrm[q].y = agg[q].y * inv * ns.y;
        nrm[q].z = agg[q].z * inv * ns.z;
        nrm[q].w = agg[q].w * inv * ns.w;
    }

    v4f* of = (v4f*)(out + (size_t)t * KK);
#pragma unroll
    for (int i = 0; i < 4; ++i) {
#pragma unroll
        for (int q = 0; q < 2; ++q) {
            v4f o;
            o.x = sHpost[i] * nrm[q].x;
            o.y = sHpost[i] * nrm[q].y;
            o.z = sHpost[i] * nrm[q].z;
            o.w = sHpost[i] * nrm[q].w;
#pragma unroll
            for (int j = 0; j < 4; ++j) o = v4fma(sM[i*4 + j], xv[j][q], o);
            __builtin_nontemporal_store(o, of + i * 512 + tid + q * 256);
        }
    }
}

// ---------------------------------------------------------------------------
extern "C" void kernel_launch(void* const* d_in, const int* in_sizes, int n_in,
                              void* d_out, int out_size, void* d_ws, size_t ws_size,
                              hipStream_t stream) {
    const float* x          = (const float*)d_in[0];
    const float* phi_pre    = (const float*)d_in[1];
    const float* phi_post   = (const float*)d_in[2];
    const float* phi_res    = (const float*)d_in[3];
    const float* b_pre      = (const float*)d_in[4];
    const float* b_post     = (const float*)d_in[5];
    const float* b_res      = (const float*)d_in[6];
    const float* alpha_pre  = (const float*)d_in[7];
    const float* alpha_post = (const float*)d_in[8];
    const float* alpha_res  = (const float*)d_in[9];
    const float* norm_scale = (const float*)d_in[10];
    float* out = (float*)d_out;

    // workspace layout (fp32): fragB | Hpre | Hpost | M  (~1.75 MB total)
    float* ws     = (float*)d_ws;
    float* fragB  = ws;                                // KK*32
    float* wHpre  = fragB + (size_t)KK * 32;           // TOK*4
    float* wHpost = wHpre + (size_t)TOK * 4;           // TOK*4
    float* wM     = wHpost + (size_t)TOK * 4;          // TOK*16

    pack_phi_kernel<<<(KK * 32) / 256, 256, 0, stream>>>(
        phi_pre, phi_post, phi_res, alpha_pre, alpha_post, alpha_res, fragB);

    gates_kernel<<<TOK / 16, KWAVES * 32, 0, stream>>>(
        x, fragB, b_pre, b_post, b_res, wHpre, wHpost, wM);

    output_kernel<<<TOK, 256, 0, stream>>>(
        x, wHpre, wHpost, wM, norm_scale, out);
}